// MOELayer_38697655337072
// MI455X (gfx1250) — compile-verified
//
#include <hip/hip_runtime.h>
#include <hip/hip_bf16.h>
#include <math.h>
#include <stdint.h>

// Problem constants (match the reference harness).
#define BB 2
#define TT 1024
#define CC 512
#define EE 8
#define FF 2048
#define BT (BB * TT)            // 2048 tokens
#define TM 16                   // tokens per FFN block (one WMMA tile of rows)
#define FCHUNK 128              // F per fused iteration (8 waves x 16 cols)
#define KC 64                   // K rows per async-staged W1 sub-chunk
#define BLOCKS_PER_E (BT / TM)  // 128

#define XS_STRIDE 516           // 512 + 4 pad  -> conflict-free A fragments
#define HS_STRIDE 132           // 128 + 4 pad
#define W1S_STRIDE 136          // 128 + 8 pad  -> 2*136 % 64 == 16: the two
                                //  half-wave row reads hit disjoint banks

typedef __attribute__((ext_vector_type(2))) float v2f;
typedef __attribute__((ext_vector_type(8))) float v8f;

// ---------------------------------------------------------------------------
// CDNA5 async copy engine: global -> LDS without touching VGPRs, tracked by
// ASYNCcnt (in-order completion). One instruction moves 16 B per lane.
// ---------------------------------------------------------------------------
__device__ __forceinline__ void async_copy_b128(uint32_t lds_off,
                                                const float* gptr) {
  asm volatile("global_load_async_to_lds_b128 %0, %1, off"
               :
               : "v"(lds_off), "v"(gptr)
               : "memory");
}
__device__ __forceinline__ void wait_async_le(int n) {
  if (n == 0)
    asm volatile("s_wait_asynccnt 0x0" ::: "memory");
  else
    asm volatile("s_wait_asynccnt 0x8" ::: "memory");
}

// ---------------------------------------------------------------------------
// Kernel 0: zero the output accumulator and the per-expert counters.
// ---------------------------------------------------------------------------
__global__ __launch_bounds__(256) void moe_zero(float* __restrict__ out, int n,
                                                int* __restrict__ cnt) {
  int i = blockIdx.x * 256 + threadIdx.x;
  if (i < n) out[i] = 0.0f;
  if (blockIdx.x == 0 && threadIdx.x < EE) cnt[threadIdx.x] = 0;
}

// ---------------------------------------------------------------------------
// Kernel 1: router. One wave per token; lanes split C, butterfly-reduce the
// 8 logits (wave32), softmax + top-2 on lane 0, append to expert lists.
// ---------------------------------------------------------------------------
__global__ __launch_bounds__(256) void moe_router(
    const float* __restrict__ X, const float* __restrict__ Wr,
    int* __restrict__ cnt, int* __restrict__ tok, float* __restrict__ wgt) {
  const int wave = threadIdx.x >> 5;
  const int lane = threadIdx.x & 31;
  const int t = blockIdx.x * 8 + wave;
  if (t >= BT) return;

  float acc[EE];
#pragma unroll
  for (int e = 0; e < EE; ++e) acc[e] = 0.0f;

  const float* xr = X + (size_t)t * CC;
  for (int c = lane; c < CC; c += 32) {
    float x = xr[c];
    const float* wr = Wr + (size_t)c * EE;
#pragma unroll
    for (int e = 0; e < EE; ++e) acc[e] += x * wr[e];
  }
#pragma unroll
  for (int off = 16; off > 0; off >>= 1) {
#pragma unroll
    for (int e = 0; e < EE; ++e) acc[e] += __shfl_xor(acc[e], off, 32);
  }

  if (lane == 0) {
    float mx = acc[0];
#pragma unroll
    for (int e = 1; e < EE; ++e) mx = fmaxf(mx, acc[e]);
    float p[EE];
    float s = 0.0f;
#pragma unroll
    for (int e = 0; e < EE; ++e) { p[e] = __expf(acc[e] - mx); s += p[e]; }
    float inv = 1.0f / s;
#pragma unroll
    for (int e = 0; e < EE; ++e) p[e] *= inv;

    int i1 = 0;
#pragma unroll
    for (int e = 1; e < EE; ++e) if (p[e] > p[i1]) i1 = e;
    int i2 = (i1 == 0) ? 1 : 0;
#pragma unroll
    for (int e = 0; e < EE; ++e)
      if (e != i1 && p[e] > p[i2]) i2 = e;

    int s1 = atomicAdd(&cnt[i1], 1);
    tok[i1 * BT + s1] = t;
    wgt[i1 * BT + s1] = p[i1];
    int s2 = atomicAdd(&cnt[i2], 1);
    tok[i2 * BT + s2] = t;
    wgt[i2 * BT + s2] = p[i2];
  }
}

// ---------------------------------------------------------------------------
// Kernel 2: fused per-expert FFN, fp32 WMMA (v_wmma_f32_16x16x4_f32).
//   - X tile (16x512) gathered into LDS via async-to-LDS b128.
//   - W1 streamed through LDS in double-buffered 64x128 sub-chunks via the
//     async copy engine (shared by all 8 waves), overlapped with WMMA.
//   - W2 streamed directly from global (no cross-wave reuse -> no staging).
//   - Epilogue: out[token] += w * (y + b2) via global_atomic_add_f32
//     (exactly 2 adds per element -> order-independent, deterministic).
// ---------------------------------------------------------------------------
__global__ __launch_bounds__(256) void moe_ffn(
    const float* __restrict__ X,
    const float* __restrict__ W1, const float* __restrict__ b1,
    const float* __restrict__ W2, const float* __restrict__ b2,
    const int* __restrict__ cnt, const int* __restrict__ tok,
    const float* __restrict__ wgt, float* __restrict__ out) {
  const int e = blockIdx.x >> 7;  // / BLOCKS_PER_E
  const int blk = blockIdx.x & (BLOCKS_PER_E - 1);
  const int n_e = cnt[e];
  const int row0 = blk * TM;
  if (row0 >= n_e) return;  // block-uniform early exit

  __shared__ __align__(16) float Xs[TM * XS_STRIDE];
  __shared__ __align__(16) float Hs[TM * HS_STRIDE];
  __shared__ __align__(16) float W1s[2][KC * W1S_STRIDE];
  __shared__ int s_tok[TM];
  __shared__ float s_wgt[TM];

  const int tid = threadIdx.x;
  if (tid < TM) {
    int r = row0 + tid;
    bool v = (r < n_e);
    s_tok[tid] = v ? tok[e * BT + r] : -1;
    s_wgt[tid] = v ? wgt[e * BT + r] : 0.0f;
  }
  __syncthreads();

  const float* W1e = W1 + (size_t)e * CC * FF;
  const float* W2e = W2 + (size_t)e * FF * CC;

  // Async W1 sub-chunk fetch: rows [kbase, kbase+KC) x cols [fc_, fc_+128)
  // -> W1s[buf].  8 b128 instructions per wave, 512 B each, fully coalesced.
  auto issue_w1 = [&](int fc_, int kbase, int buf) {
#pragma unroll
    for (int t = 0; t < 8; ++t) {
      int i = tid + (t << 8);       // 0..2047
      int row = i >> 5;             // 32 float4 per 128-wide row
      int c4 = (i & 31) << 2;
      const float* g = W1e + (size_t)(kbase + row) * FF + fc_ + c4;
      async_copy_b128((uint32_t)(uintptr_t)&W1s[buf][row * W1S_STRIDE + c4],
                      g);
    }
  };

  // Gather X tile [16 x 512] into LDS via async copies (zero-fill invalid
  // rows so the WMMA region runs with full EXEC and garbage-free math).
#pragma unroll
  for (int t = 0; t < 8; ++t) {
    int i = tid + (t << 8);         // 0..2047
    int r = i >> 7;                 // 128 float4 per 512-wide row
    int c4 = (i & 127) << 2;
    int tkn = s_tok[r];
    if (tkn >= 0) {
      async_copy_b128((uint32_t)(uintptr_t)&Xs[r * XS_STRIDE + c4],
                      X + (size_t)tkn * CC + c4);
    } else {
      *(float4*)&Xs[r * XS_STRIDE + c4] = make_float4(0.f, 0.f, 0.f, 0.f);
    }
  }
  // Prefetch the first W1 sub-chunk (fc=0, k rows 0..63) into buffer 0.
  issue_w1(0, 0, 0);

  const int wv = tid >> 5;
  const int lane = tid & 31;
  const int m = lane & 15;   // row (A/D) or column (B/D) index
  const int half = lane >> 4;

  v8f accy[4];
#pragma unroll
  for (int j = 0; j < 4; ++j)
#pragma unroll
    for (int i = 0; i < 8; ++i) accy[j][i] = 0.0f;

  const int n0base = wv * 64;  // this wave's y-column slice
  const float* Xrow = &Xs[m * XS_STRIDE + 2 * half];

  for (int fc = 0; fc < FF; fc += FCHUNK) {
    // ---- stage A: h tile [16 tokens x 16 f-cols] per wave, K swept in
    //      8 async-staged sub-chunks of 64 with double buffering ----
    const int f0 = fc + wv * 16;
    const float bval = b1[e * FF + f0 + m];
    v8f acc;
#pragma unroll
    for (int i = 0; i < 8; ++i) acc[i] = bval;

    for (int kc = 0; kc < 8; ++kc) {
      bool issued = true;
      if (kc < 7)
        issue_w1(fc, (kc + 1) * KC, (kc + 1) & 1);       // next sub-chunk
      else if (fc + FCHUNK < FF)
        issue_w1(fc + FCHUNK, 0, 0);                      // next f-chunk
      else
        issued = false;
      wait_async_le(issued ? 8 : 0);  // retire older batch only -> overlap
      __syncthreads();

      const float* Ar = Xrow + kc * KC;
      const float* Bb = &W1s[kc & 1][(2 * half) * W1S_STRIDE + wv * 16 + m];
#pragma unroll
      for (int k0 = 0; k0 < KC; k0 += 4) {
        v2f a;
        a.x = Ar[k0];
        a.y = Ar[k0 + 1];
        v2f b;
        b.x = Bb[k0 * W1S_STRIDE];
        b.y = Bb[k0 * W1S_STRIDE + W1S_STRIDE];
        acc = __builtin_amdgcn_wmma_f32_16x16x4_f32(
            false, a, false, b, (short)0, acc, false, false);
      }
    }
    // relu + store h chunk to LDS
#pragma unroll
    for (int i = 0; i < 8; ++i) {
      float h = acc[i] > 0.0f ? acc[i] : 0.0f;
      Hs[(i + 8 * half) * HS_STRIDE + wv * 16 + m] = h;
    }
    __syncthreads();

    // ---- stage B: y[16 x 64] += h_chunk @ W2[fc:fc+128, cols] ----
    // (overlaps with the async prefetch of the next f-chunk's W1)
    const float* Hrow = &Hs[m * HS_STRIDE + 2 * half];
#pragma unroll
    for (int j = 0; j < 4; ++j) {
      const int n0 = n0base + j * 16;
      const float* B2p = W2e + (size_t)(fc + 2 * half) * CC + n0 + m;
      v8f ay = accy[j];
#pragma unroll 8
      for (int kk = 0; kk < FCHUNK; kk += 4) {
        v2f a;
        a.x = Hrow[kk];
        a.y = Hrow[kk + 1];
        v2f b;
        b.x = B2p[(size_t)kk * CC];
        b.y = B2p[(size_t)(kk + 1) * CC];
        ay = __builtin_amdgcn_wmma_f32_16x16x4_f32(
            false, a, false, b, (short)0, ay, false, false);
      }
      accy[j] = ay;
    }
    // no trailing barrier: the next f-chunk's per-sub-chunk barriers already
    // order all waves' Hs reads before Hs is overwritten.
  }

  // ---- epilogue: weighted scatter-add ----
#pragma unroll
  for (int j = 0; j < 4; ++j) {
    const int c = n0base + j * 16 + m;
    const float b2v = b2[e * CC + c];
#pragma unroll
    for (int i = 0; i < 8; ++i) {
      int r = i + 8 * half;
      int tkn = s_tok[r];
      if (tkn >= 0) {
        float val = s_wgt[r] * (accy[j][i] + b2v);
        atomicAdd(out + (size_t)tkn * CC + c, val);
      }
    }
  }
}

// ---------------------------------------------------------------------------
// Launch: zero -> route -> fused expert FFN. Workspace layout:
//   [0, 32)            : int cnt[8]
//   [256, 256+64K)     : int tok[8][2048]
//   [256+64K, +64K)    : float wgt[8][2048]
// ---------------------------------------------------------------------------
extern "C" void kernel_launch(void* const* d_in, const int* in_sizes, int n_in,
                              void* d_out, int out_size, void* d_ws,
                              size_t ws_size, hipStream_t stream) {
  const float* X  = (const float*)d_in[0];
  const float* Wr = (const float*)d_in[1];
  const float* W1 = (const float*)d_in[2];
  const float* b1 = (const float*)d_in[3];
  const float* W2 = (const float*)d_in[4];
  const float* b2 = (const float*)d_in[5];
  // d_in[6] = top_k (fixed at 2 by the reference).
  float* out = (float*)d_out;

  char* ws = (char*)d_ws;
  int* cnt   = (int*)ws;
  int* tok   = (int*)(ws + 256);
  float* wgt = (float*)(ws + 256 + (size_t)EE * BT * sizeof(int));

  moe_zero<<<(out_size + 255) / 256, 256, 0, stream>>>(out, out_size, cnt);
  moe_router<<<BT / 8, 256, 0, stream>>>(X, Wr, cnt, tok, wgt);
  moe_ffn<<<EE * BLOCKS_PER_E, 256, 0, stream>>>(X, W1, b1, W2, b2, cnt, tok,
                                                 wgt, out);
}